// SparseMaskGenerator_40269613367471
// MI455X (gfx1250) — compile-verified
//
#include <hip/hip_runtime.h>
#include <stdint.h>

// Problem constants (match reference): B=32, S=2048, F=512
#define FEAT 512
#define ROWS_TOTAL (32 * 2048)      // 65536 rows of length F
#define RPW 8                       // rows processed per wave
#define WPB 8                       // waves per block (256 threads)
#define ROWBUF_BYTES 4096           // 2048B logits + 2048B gumbel
#define WAVE_LDS (2 * ROWBUF_BYTES) // double buffered = 8 KB per wave

typedef float v4f __attribute__((ext_vector_type(4)));

// Stage one row (512 floats of logits + 512 floats of gumbel) into LDS with
// CDNA5 async global->LDS copies. Each of the 8 instructions moves 16 B/lane
// (512 B/wave); the inst offset applies to BOTH the global and LDS address,
// so the LDS buffer mirrors the global row layout. Tracked by ASYNCcnt.
// th:TH_LOAD_NT: single-pass stream, don't let it occupy L2.
__device__ __forceinline__ void async_stage_row(uint32_t lds_logits, uint32_t lds_gumbel,
                                                const float* __restrict__ gl,
                                                const float* __restrict__ gg, int lane) {
  uint64_t al = (uint64_t)(uintptr_t)gl + (uint64_t)lane * 16;
  uint64_t ag = (uint64_t)(uintptr_t)gg + (uint64_t)lane * 16;
  uint32_t dl = lds_logits + lane * 16;
  uint32_t dg = lds_gumbel + lane * 16;
  asm volatile(
      "global_load_async_to_lds_b128 %0, %1, off th:TH_LOAD_NT\n\t"
      "global_load_async_to_lds_b128 %0, %1, off offset:512 th:TH_LOAD_NT\n\t"
      "global_load_async_to_lds_b128 %0, %1, off offset:1024 th:TH_LOAD_NT\n\t"
      "global_load_async_to_lds_b128 %0, %1, off offset:1536 th:TH_LOAD_NT\n\t"
      "global_load_async_to_lds_b128 %2, %3, off th:TH_LOAD_NT\n\t"
      "global_load_async_to_lds_b128 %2, %3, off offset:512 th:TH_LOAD_NT\n\t"
      "global_load_async_to_lds_b128 %2, %3, off offset:1024 th:TH_LOAD_NT\n\t"
      "global_load_async_to_lds_b128 %2, %3, off offset:1536 th:TH_LOAD_NT"
      :: "v"(dl), "v"(al), "v"(dg), "v"(ag)
      : "memory");
}

__global__ __launch_bounds__(WPB * 32) void SparseMaskGenerator_40269613367471_kernel(
    const float* __restrict__ logits, const float* __restrict__ gumbel,
    float* __restrict__ out) {
  __shared__ alignas(16) char smem[WPB * WAVE_LDS]; // 64 KB

  const int lane = threadIdx.x & 31;
  const int wave = threadIdx.x >> 5;
  const int gwave = blockIdx.x * WPB + wave;
  const int row0 = gwave * RPW;

  char* wbase = smem + wave * WAVE_LDS;
  // Low 32 bits of a generic shared pointer are the workgroup-relative LDS
  // byte offset; the async unit adds LDS_BASE itself.
  const uint32_t wbase_u = (uint32_t)(uintptr_t)wbase;

  // Incremental row pointers (one 64-bit add per row instead of re-deriving).
  const float* lrow = logits + (size_t)row0 * FEAT;
  const float* grow = gumbel + (size_t)row0 * FEAT;
  float* orow = out + (size_t)row0 * FEAT;

  // Prologue: prefetch row 0 into buffer 0 (asynccnt -> 8)
  async_stage_row(wbase_u, wbase_u + 2048, lrow, grow, lane);

  for (int r = 0; r < RPW; ++r) {
    const int buf = r & 1;
    const int nbuf = buf ^ 1;
    if (r + 1 < RPW) {
      // Prefetch next row into the other buffer, then wait until only those
      // 8 copies are outstanding: the current row's 8 (issued first, retired
      // in order) are then guaranteed resident in LDS.
      async_stage_row(wbase_u + nbuf * ROWBUF_BYTES,
                      wbase_u + nbuf * ROWBUF_BYTES + 2048,
                      lrow + FEAT, grow + FEAT, lane);
      asm volatile("s_wait_asynccnt 0x8" ::: "memory");
    } else {
      asm volatile("s_wait_asynccnt 0x0" ::: "memory");
    }

    // Lane l owns features [16l, 16l+16): 4x ds_load_b128 per operand.
    const v4f* Lv = (const v4f*)(wbase + buf * ROWBUF_BYTES + lane * 64);
    const v4f* Gv = (const v4f*)(wbase + buf * ROWBUF_BYTES + 2048 + lane * 64);

    float z[16];
#pragma unroll
    for (int k = 0; k < 4; ++k) {
      v4f a = Lv[k];
      v4f b = Gv[k];
      z[4 * k + 0] = a.x + b.x;
      z[4 * k + 1] = a.y + b.y;
      z[4 * k + 2] = a.z + b.z;
      z[4 * k + 3] = a.w + b.w;
    }

    // Lane-local argmax (first-max tie-break), then wave32 reduction.
    float m = z[0];
    int mi = 0;
#pragma unroll
    for (int j = 1; j < 16; ++j)
      if (z[j] > m) { m = z[j]; mi = j; }
    int gidx = lane * 16 + mi;

#pragma unroll
    for (int off = 16; off > 0; off >>= 1) {
      float om = __shfl_xor(m, off, 32);
      int og = __shfl_xor(gidx, off, 32);
      if (om > m || (om == m && og < gidx)) { m = om; gidx = og; }
    }

    // Softmax denominator: sum of exp(z - max); the max element contributes
    // exp(0)=1, so p = y_soft[argmax] = 1/s.
    float s = 0.0f;
#pragma unroll
    for (int j = 0; j < 16; ++j) s += __expf(z[j] - m);
#pragma unroll
    for (int off = 16; off > 0; off >>= 1) s += __shfl_xor(s, off, 32);

    const float p = 1.0f / s;
    const float val = (1.0f - p) + p; // straight-through value at argmax

    // Zero-fill the row with non-temporal b128 stores, then the single lane
    // owning the argmax overwrites one dword. Same-wave stores to the same
    // address retire in program order (ISA: "stores in order with stores"),
    // so the scalar store lands last.
    v4f* O = (v4f*)(orow + lane * 16);
    const v4f zv = {0.0f, 0.0f, 0.0f, 0.0f};
#pragma unroll
    for (int k = 0; k < 4; ++k) __builtin_nontemporal_store(zv, O + k);
    if ((gidx >> 4) == lane) orow[gidx] = val;

    lrow += FEAT;
    grow += FEAT;
    orow += FEAT;
  }
}

extern "C" void kernel_launch(void* const* d_in, const int* in_sizes, int n_in,
                              void* d_out, int out_size, void* d_ws, size_t ws_size,
                              hipStream_t stream) {
  (void)in_sizes; (void)n_in; (void)d_ws; (void)ws_size; (void)out_size;
  const float* logits = (const float*)d_in[0];
  const float* gumbel = (const float*)d_in[1];
  float* out = (float*)d_out;

  const int blocks = ROWS_TOTAL / (RPW * WPB); // 65536 / 64 = 1024
  SparseMaskGenerator_40269613367471_kernel<<<blocks, WPB * 32, 0, stream>>>(
      logits, gumbel, out);
}